// GNN_31525059953012
// MI455X (gfx1250) — compile-verified
//
#include <hip/hip_runtime.h>

#define N_NODES 50000
#define E_EDGES 625000
#define D_FEAT  128

typedef __attribute__((ext_vector_type(2))) float v2f;
typedef __attribute__((ext_vector_type(8))) float v8f;

// ---------------------------------------------------------------------------
// Zero the (sum, cnt) accumulators in workspace.
// ---------------------------------------------------------------------------
__global__ __launch_bounds__(256) void zero_kernel(float* __restrict__ p, int n) {
    int i = blockIdx.x * 256 + threadIdx.x;
    if (i < n) p[i] = 0.0f;
}

// ---------------------------------------------------------------------------
// Edge scatter: one wave32 per edge. Lanes read one float4 each (coalesced
// 512B gather of x[src]), then 4 global_atomic_add_f32 into sum[dst].
// Lane 0 bumps the degree counter. Entire working set is L2-resident
// (x: 25.6MB, sum: 25.6MB, L2: 192MB) so atomics resolve in L2.
// ---------------------------------------------------------------------------
__global__ __launch_bounds__(256) void scatter_kernel(
    const float* __restrict__ x,
    const long long* __restrict__ src,
    const long long* __restrict__ dst,
    float* __restrict__ sum,
    float* __restrict__ cnt,
    int nEdges)
{
    int gid  = blockIdx.x * 256 + threadIdx.x;
    int e    = gid >> 5;          // wave32: one edge per wave
    int lane = gid & 31;
    if (e >= nEdges) return;

    long long s = src[e];
    long long d = dst[e];

    float4 v = ((const float4*)(x + s * (long long)D_FEAT))[lane];
    float* o = sum + d * (long long)D_FEAT + lane * 4;
    atomicAdd(o + 0, v.x);
    atomicAdd(o + 1, v.y);
    atomicAdd(o + 2, v.z);
    atomicAdd(o + 3, v.w);
    if (lane == 0) atomicAdd(cnt + d, 1.0f);
}

// ---------------------------------------------------------------------------
// Fused SAGE linear stage: out[i,:] = mean[i,:] @ Wl^T + bl + xin[i,:] @ Wr^T
// (optional ReLU). Block = 256 threads = 8 waves; block owns 16 node rows,
// wave w owns output columns [16w, 16w+16). fp32 WMMA 16x16x4, K-loop of 32
// steps per matrix. mean is formed on the fly from (sum, cnt) while staging
// into LDS (rows padded to 132 floats to avoid DS bank conflicts).
// Uniform control flow everywhere -> EXEC all-ones at every WMMA.
// ---------------------------------------------------------------------------
template <bool RELU>
__global__ __launch_bounds__(256) void sage_gemm_kernel(
    const float* __restrict__ xin,
    const float* __restrict__ sum,
    const float* __restrict__ cnt,
    const float* __restrict__ Wl,
    const float* __restrict__ bl,
    const float* __restrict__ Wr,
    float* __restrict__ out)
{
    __shared__ float lds_m[16][132];   // mean tile  (16 rows x 128, padded)
    __shared__ float lds_x[16][132];   // x/h tile

    const int tid  = threadIdx.x;
    const int row0 = blockIdx.x * 16;

    // ---- stage 16x128 tiles of mean and x into LDS (float4 transfers) ----
    #pragma unroll
    for (int c = tid; c < 512; c += 256) {      // 512 float4 chunks per array
        int row = c >> 5;                       // 32 float4 chunks per row
        int col = (c & 31) * 4;
        int g   = (row0 + row) * D_FEAT + col;
        float4 xv = *(const float4*)(xin + g);
        float4 sv = *(const float4*)(sum + g);
        float  inv = 1.0f / fmaxf(cnt[row0 + row], 1.0f);
        *(float4*)&lds_x[row][col] = xv;
        float4 mv;
        mv.x = sv.x * inv; mv.y = sv.y * inv; mv.z = sv.z * inv; mv.w = sv.w * inv;
        *(float4*)&lds_m[row][col] = mv;
    }
    __syncthreads();

    // ---- WMMA fragment indexing (32-bit operands, ISA 7.12.2) ----
    const int wave = tid >> 5;                 // 0..7 -> column tile
    const int lane = tid & 31;
    const int m    = lane & 15;                // A: row M / B: col N
    const int koff = (lane < 16) ? 0 : 2;      // K half per lane group
    const int ncol = wave * 16 + m;            // global output column

    const float* wl_row = Wl + ncol * D_FEAT;  // W[n][k], contiguous in k
    const float* wr_row = Wr + ncol * D_FEAT;

    v8f acc;
    {
        float bias = bl[ncol];                 // bias depends only on N
        #pragma unroll
        for (int i = 0; i < 8; ++i) acc[i] = bias;
    }

    #pragma unroll 4
    for (int k = 0; k < D_FEAT; k += 4) {
        // mean @ Wl^T
        v2f a0 = *(const v2f*)&lds_m[m][k + koff];
        v2f b0 = *(const v2f*)(wl_row + k + koff);
        acc = __builtin_amdgcn_wmma_f32_16x16x4_f32(
                  false, a0, false, b0, (short)0, acc, false, false);
        // x @ Wr^T
        v2f a1 = *(const v2f*)&lds_x[m][k + koff];
        v2f b1 = *(const v2f*)(wr_row + k + koff);
        acc = __builtin_amdgcn_wmma_f32_16x16x4_f32(
                  false, a1, false, b1, (short)0, acc, false, false);
    }

    // ---- store D tile: VGPR v, lane l -> row = v + (l<16?0:8), col = ncol ----
    const int rbase = row0 + ((lane < 16) ? 0 : 8);
    #pragma unroll
    for (int v = 0; v < 8; ++v) {
        float r = acc[v];
        if (RELU) r = fmaxf(r, 0.0f);
        out[(rbase + v) * D_FEAT + ncol] = r;
    }
}

// ---------------------------------------------------------------------------
// Host-side launch.  Inputs (setup_inputs order):
//  0 x[N,D] f32 | 1 edge_index[2,E] i64 | 2 W1_l | 3 b1_l | 4 W1_r
//  5 W2_l | 6 b2_l | 7 W2_r.  Output: [N,D] f32.
// Workspace: sum[N*D] f32 + cnt[N] f32 (~25.8MB).
// Layer-1 activations (post-ReLU) live in d_out; layer 2 reads its own rows
// from d_out before overwriting them (row-local, block-exclusive -> safe).
// ---------------------------------------------------------------------------
extern "C" void kernel_launch(void* const* d_in, const int* in_sizes, int n_in,
                              void* d_out, int out_size, void* d_ws, size_t ws_size,
                              hipStream_t stream)
{
    const float*     x   = (const float*)d_in[0];
    const long long* ei  = (const long long*)d_in[1];
    const float*     W1l = (const float*)d_in[2];
    const float*     b1l = (const float*)d_in[3];
    const float*     W1r = (const float*)d_in[4];
    const float*     W2l = (const float*)d_in[5];
    const float*     b2l = (const float*)d_in[6];
    const float*     W2r = (const float*)d_in[7];
    float*           out = (float*)d_out;

    const long long* src = ei;             // edge_index[0]
    const long long* dst = ei + E_EDGES;   // edge_index[1]

    float* sum = (float*)d_ws;
    float* cnt = sum + (size_t)N_NODES * D_FEAT;

    const int totalZero   = N_NODES * D_FEAT + N_NODES;
    const int zeroBlocks  = (totalZero + 255) / 256;
    const int scatBlocks  = (E_EDGES * 32 + 255) / 256;   // one wave per edge
    const int gemmBlocks  = N_NODES / 16;                 // 50000/16 = 3125
    dim3 blk(256);

    // ---- layer 1 ----
    zero_kernel<<<zeroBlocks, blk, 0, stream>>>(sum, totalZero);
    scatter_kernel<<<scatBlocks, blk, 0, stream>>>(x, src, dst, sum, cnt, E_EDGES);
    sage_gemm_kernel<true><<<gemmBlocks, blk, 0, stream>>>(x, sum, cnt, W1l, b1l, W1r, out);

    // ---- layer 2 ----
    zero_kernel<<<zeroBlocks, blk, 0, stream>>>(sum, totalZero);
    scatter_kernel<<<scatBlocks, blk, 0, stream>>>(out, src, dst, sum, cnt, E_EDGES);
    sage_gemm_kernel<false><<<gemmBlocks, blk, 0, stream>>>(out, sum, cnt, W2l, b2l, W2r, out);
}